// CTCLayer_9053791060176
// MI455X (gfx1250) — compile-verified
//
#include <hip/hip_runtime.h>
#include <hip/hip_bf16.h>
#include <math.h>

// CTC batch cost, shapes fixed by the reference: B=64, T=128, C=4000, L=48.
#define CTC_B     64
#define CTC_T     128
#define CTC_C     4000
#define CTC_L     48
#define CTC_S     (2 * CTC_L + 1)   // 97 extended states
#define CTC_SP    104               // padded to 26 lanes * 4 states (16B groups)
#define CTC_BLANK (CTC_C - 1)
#define NEGV      (-1e30f)
#define EPSV      (1e-7f)
#define NTHREADS  256
#define DP_LANES  (CTC_SP / 4)      // 26 lanes carry real/padded states

__device__ __forceinline__ float lse3(float x, float y, float z) {
  float m = fmaxf(fmaxf(x, y), z);
  return m + logf(expf(x - m) + expf(y - m) + expf(z - m));
}

__global__ __launch_bounds__(NTHREADS) void ctc_loss_kernel(
    const int*   __restrict__ y_true,     // [B, L]
    const float* __restrict__ y_pred,     // [B, T, C] probabilities
    const int*   __restrict__ label_len,  // [B, 1]
    float*       __restrict__ out)        // [B]
{
  __shared__ __align__(16) float lp[CTC_T * CTC_SP];  // log-probs at ext states
  __shared__ int   ext[CTC_SP];
  __shared__ float alphaBuf[CTC_SP];

  const int b   = blockIdx.x;
  const int tid = threadIdx.x;
  const int ll  = label_len[b];

  // Extended label sequence: blanks interleaved; labels past ll -> blank.
  if (tid < CTC_SP) {
    int e = CTC_BLANK;
    if (tid & 1) {
      int i = tid >> 1;
      if (i < ll) e = y_true[b * CTC_L + i];
    }
    ext[tid] = e;
  }
  __syncthreads();

  const float* yp = y_pred + (size_t)b * CTC_T * CTC_C;

  // ---- Gather phase: lp[t*SP + s] = log(y_pred[b,t,ext[s]] + eps) ----
#if defined(__gfx1250__) && __has_builtin(__builtin_amdgcn_global_load_async_to_lds_b32)
  typedef __attribute__((address_space(1))) int gint_t;  // global int
  typedef __attribute__((address_space(3))) int lint_t;  // LDS int
  for (int i = tid; i < CTC_T * CTC_SP; i += NTHREADS) {
    int t = i / CTC_SP;
    int s = i - t * CTC_SP;
    const float* g = yp + (size_t)t * CTC_C + ext[s];
    __builtin_amdgcn_global_load_async_to_lds_b32(
        (gint_t*)g, (lint_t*)(lp + i), 0, 0);
  }
#if __has_builtin(__builtin_amdgcn_s_wait_asynccnt)
  __builtin_amdgcn_s_wait_asynccnt(0);
#else
  asm volatile("s_wait_asynccnt 0" ::: "memory");
#endif
  asm volatile("" ::: "memory");
  for (int i = tid; i < CTC_T * CTC_SP; i += NTHREADS) {
    lp[i] = logf(lp[i] + EPSV);
  }
#else
  for (int i = tid; i < CTC_T * CTC_SP; i += NTHREADS) {
    int t = i / CTC_SP;
    int s = i - t * CTC_SP;
    lp[i] = logf(yp[(size_t)t * CTC_C + ext[s]] + EPSV);
  }
#endif
  __syncthreads();

  // ---- DP phase: single wave32, 4 consecutive states per lane ----
  if (tid < 32) {
    const int lane = tid;

    bool sk[4];
#pragma unroll
    for (int i = 0; i < 4; ++i) {
      int s = lane * 4 + i;
      if (s < CTC_SP) {
        int e   = ext[s];
        int em2 = (s >= 2) ? ext[s - 2] : CTC_BLANK;
        sk[i] = (e != CTC_BLANK) && (e != em2);
      } else {
        sk[i] = false;
      }
    }

    // alpha at t=0: only states 0 and 1 reachable.
    float4 l0 = make_float4(0.f, 0.f, 0.f, 0.f);
    if (lane < DP_LANES) l0 = reinterpret_cast<const float4*>(lp)[lane];
    float a0 = (lane == 0) ? l0.x : NEGV;
    float a1 = (lane == 0) ? l0.y : NEGV;
    float a2 = NEGV;
    float a3 = NEGV;

    for (int t = 1; t < CTC_T; ++t) {
      // Neighbor lane's alpha[s-1] (its slot 3) and alpha[s-2] (its slot 2).
      float p3 = __shfl_up(a3, 1, 32);
      float p2 = __shfl_up(a2, 1, 32);
      if (lane == 0) { p3 = NEGV; p2 = NEGV; }

      float4 l4 = make_float4(0.f, 0.f, 0.f, 0.f);
      if (lane < DP_LANES)
        l4 = reinterpret_cast<const float4*>(lp + t * CTC_SP)[lane];

      float n0 = lse3(a0, p3, sk[0] ? p2 : NEGV) + l4.x;
      float n1 = lse3(a1, a0, sk[1] ? p3 : NEGV) + l4.y;
      float n2 = lse3(a2, a1, sk[2] ? a0 : NEGV) + l4.z;
      float n3 = lse3(a3, a2, sk[3] ? a1 : NEGV) + l4.w;
      a0 = n0; a1 = n1; a2 = n2; a3 = n3;
    }

    if (lane < DP_LANES) {
      alphaBuf[lane * 4 + 0] = a0;
      alphaBuf[lane * 4 + 1] = a1;
      alphaBuf[lane * 4 + 2] = a2;
      alphaBuf[lane * 4 + 3] = a3;
    }
#if __has_builtin(__builtin_amdgcn_wave_barrier)
    __builtin_amdgcn_wave_barrier();
#endif
    asm volatile("" ::: "memory");  // same-wave LDS is in-order in HW

    if (lane == 0) {
      float ab = alphaBuf[2 * ll];       // end in final blank
      float al = alphaBuf[2 * ll - 1];   // end in final label
      float m  = fmaxf(ab, al);
      out[b] = -(m + logf(expf(ab - m) + expf(al - m)));
    }
  }
}

extern "C" void kernel_launch(void* const* d_in, const int* in_sizes, int n_in,
                              void* d_out, int out_size, void* d_ws, size_t ws_size,
                              hipStream_t stream) {
  (void)n_in; (void)d_ws; (void)ws_size; (void)out_size;
  const int*   y_true    = (const int*)d_in[0];    // [B, L] int32
  const float* y_pred    = (const float*)d_in[1];  // [B, T, C] float32
  const int*   label_len = (const int*)d_in[2];    // [B, 1] int32
  float*       out       = (float*)d_out;          // [B, 1] float32

  const int B = in_sizes[2];  // label_length count == B
  ctc_loss_kernel<<<dim3(B), dim3(NTHREADS), 0, stream>>>(y_true, y_pred,
                                                          label_len, out);
}